// NUFFTSelfadjoint_35270271435291
// MI455X (gfx1250) — compile-verified
//
#include <hip/hip_runtime.h>

// ---------------------------------------------------------------------------
// Toeplitz self-adjoint NUFFT for MI455X (gfx1250, wave32).
//   image[5,16,384,384] -> pad -> cFFT2(768) -> K[5,5] pixelwise couple (WMMA)
//   -> icFFT2 -> crop -> out[2,5,16,384,384]
// Bandwidth-bound (~2.5 GB traffic vs ~11 GFLOP): fp32 throughout.
// CDNA5 paths used:
//   * v_wmma_f32_16x16x4_f32 for the 5x5 subspace coupling (8 per pixel)
//   * global_load_async_to_lds_b64 / global_store_async_from_lds_b64 for all
//     FFT tile staging (ASYNCcnt + s_wait_asynccnt), no VGPR round-trip
//   * centered-FFT parity signs and ortho scale folded into Stockham stages
// ---------------------------------------------------------------------------

#define OSN     768
#define NN      384
#define PAD     192
#define NCOEFF  5
#define NCOIL   16
#define NCH     (NCOEFF * NCOIL)          // 80 channels
#define FFTSCALE 0.03608439182435161f     // 1/sqrt(768) (ortho, per axis)

typedef float v2f __attribute__((ext_vector_type(2)));
typedef float v8f __attribute__((ext_vector_type(8)));

__device__ __forceinline__ float2 cadd(float2 a, float2 b) { return make_float2(a.x + b.x, a.y + b.y); }
__device__ __forceinline__ float2 csub(float2 a, float2 b) { return make_float2(a.x - b.x, a.y - b.y); }
__device__ __forceinline__ float2 cmul(float2 a, float2 b) { return make_float2(a.x * b.x - a.y * b.y, a.x * b.y + a.y * b.x); }
__device__ __forceinline__ float2 cscale(float2 a, float s) { return make_float2(a.x * s, a.y * s); }

// --- CDNA5 async global<->LDS copies (ASYNCcnt), per CDNA5 ISA -----------
__device__ __forceinline__ void async_load_b64(void* lds_dst, const void* gsrc)
{
    asm volatile("global_load_async_to_lds_b64 %0, %1, off"
                 :: "v"((unsigned)(uintptr_t)lds_dst),          // LDS offset (addr[31:0])
                    "v"((unsigned long long)(uintptr_t)gsrc)
                 : "memory");
}
__device__ __forceinline__ void async_store_b64(void* gdst, const void* lds_src)
{
    asm volatile("global_store_async_from_lds_b64 %0, %1, off"
                 :: "v"((unsigned long long)(uintptr_t)gdst),
                    "v"((unsigned)(uintptr_t)lds_src)
                 : "memory");
}
__device__ __forceinline__ void wait_async0()
{
    asm volatile("s_wait_asynccnt 0x0" ::: "memory");
}

// ---------------------------------------------------------------------------
// Mixed-radix Stockham FFT, N=768 = 3 * 4^4, in LDS.  DIR=-1 fwd, +1 inv.
// ncol independent columns laid out [col*768 + idx].  Returns result buffer.
// SIGN_IN : multiply input sample n by (-1)^n, folded into stage 0
//           (valid: stage-0 gathers t, t+256, t+512 share parity of t).
// SIGN_OUT: multiply output bin k by (-1)^k * 1/sqrt(768), folded into the
//           last stage (output index = blk*768 + k + 192*e -> parity = k&1).
// ---------------------------------------------------------------------------
template <int DIR, bool SIGN_IN, bool SIGN_OUT>
__device__ float2* fft768(float2* bufA, float2* bufB, int ncol, int tid, int nthr)
{
    float2* src = bufA;
    float2* dst = bufB;
    int Ns = 1;
#pragma unroll
    for (int stage = 0; stage < 5; ++stage) {
        const int R   = (stage == 0) ? 3 : 4;
        const int nbf = OSN / R;                     // 256 or 192 butterflies/col
        for (int i = tid; i < ncol * nbf; i += nthr) {
            const int col = i / nbf;
            const int t   = i - col * nbf;
            const int k   = t % Ns;
            const int blk = t / Ns;
            const float2* S  = src + col * OSN;
            float2*       Dd = dst + col * OSN;

            float2 x0 = S[t];
            float2 x1 = S[t + nbf];
            float2 x2 = S[t + 2 * nbf];
            if (SIGN_IN && stage == 0 && (t & 1)) {  // centered-FFT input parity
                x0 = cscale(x0, -1.f); x1 = cscale(x1, -1.f); x2 = cscale(x2, -1.f);
            }

            const float ang = (float)DIR * 6.28318530717958647f * (float)k / (float)(Ns * R);
            float s1, c1, s2, c2;
            __sincosf(ang, &s1, &c1);
            __sincosf(2.0f * ang, &s2, &c2);
            x1 = cmul(x1, make_float2(c1, s1));
            x2 = cmul(x2, make_float2(c2, s2));

            const int ob = blk * Ns * R + k;
            if (R == 3) {
                const float s3 = (float)DIR * 0.86602540378443865f;  // sin(DIR*2pi/3)
                float2 t1 = cadd(x1, x2);
                float2 t2 = csub(x1, x2);
                float2 m  = make_float2(x0.x - 0.5f * t1.x, x0.y - 0.5f * t1.y);
                float2 jt = make_float2(-s3 * t2.y, s3 * t2.x);      // i*s3*(x1-x2)
                Dd[ob]          = cadd(x0, t1);
                Dd[ob + Ns]     = cadd(m, jt);
                Dd[ob + 2 * Ns] = csub(m, jt);
            } else {
                float2 x3 = S[t + 3 * nbf];
                float s3v, c3;
                __sincosf(3.0f * ang, &s3v, &c3);
                x3 = cmul(x3, make_float2(c3, s3v));
                float2 a0 = cadd(x0, x2);
                float2 a1 = csub(x0, x2);
                float2 a2 = cadd(x1, x3);
                float2 a3 = csub(x1, x3);
                // DIR*i*a3
                float2 ja3 = (DIR < 0) ? make_float2(a3.y, -a3.x) : make_float2(-a3.y, a3.x);
                float2 y0 = cadd(a0, a2);
                float2 y1 = cadd(a1, ja3);
                float2 y2 = csub(a0, a2);
                float2 y3 = csub(a1, ja3);
                if (SIGN_OUT && stage == 4) {        // centered-FFT output parity + ortho
                    const float s = (k & 1) ? -FFTSCALE : FFTSCALE;
                    y0 = cscale(y0, s); y1 = cscale(y1, s);
                    y2 = cscale(y2, s); y3 = cscale(y3, s);
                }
                Dd[ob]          = y0;
                Dd[ob + Ns]     = y1;
                Dd[ob + 2 * Ns] = y2;
                Dd[ob + 3 * Ns] = y3;
            }
        }
        __syncthreads();
        float2* tmp = src; src = dst; dst = tmp;
        Ns *= R;
    }
    return src;   // after final swap, src holds the result
}

// ---------------------------------------------------------------------------
// Pass 1: centered pad + forward row FFT.  grid[ch][y][x], ch = a*16+c.
// Input sign applied during padded staging; output sign/scale folded in FFT;
// result streamed out LDS->global with async stores.
// ---------------------------------------------------------------------------
__global__ __launch_bounds__(256) void fwd_row_fft(const float* __restrict__ imr,
                                                   const float* __restrict__ imi,
                                                   float2* __restrict__ grid)
{
    __shared__ float2 bufA[OSN];
    __shared__ float2 bufB[OSN];
    const int y   = blockIdx.x;
    const int ch  = blockIdx.y;
    const int tid = threadIdx.x;
    float2* row = grid + (size_t)(ch * OSN + y) * OSN;

    if (y < PAD || y >= PAD + NN) {                 // zero-padded rows
        for (int x = tid; x < OSN; x += 256) row[x] = make_float2(0.f, 0.f);
        return;
    }
    const size_t ibase = ((size_t)ch * NN + (y - PAD)) * NN;
    for (int x = tid; x < OSN; x += 256) {
        float2 v = make_float2(0.f, 0.f);
        if (x >= PAD && x < PAD + NN) {
            const float sgn = (x & 1) ? -1.f : 1.f;
            v.x = sgn * imr[ibase + (x - PAD)];
            v.y = sgn * imi[ibase + (x - PAD)];
        }
        bufA[x] = v;
    }
    __syncthreads();
    float2* res = fft768<-1, false, true>(bufA, bufB, 1, tid, 256);
    for (int k = tid; k < OSN; k += 256)
        async_store_b64(&row[k], &res[k]);          // LDS -> global, no VGPR
}

// ---------------------------------------------------------------------------
// Pass 2/4: column FFT over y, 4 columns per block, in place.
// Tile staged global->LDS with per-lane async copies (transpose into
// [col*768+y] layout via the LDS destination address), signs folded in FFT,
// async store back.
// ---------------------------------------------------------------------------
template <int DIR>
__global__ __launch_bounds__(256) void col_fft(float2* __restrict__ grid)
{
    __shared__ float2 bufA[4 * OSN];
    __shared__ float2 bufB[4 * OSN];
    const int x0  = blockIdx.x * 4;
    const int ch  = blockIdx.y;
    const int tid = threadIdx.x;
    float2* base = grid + (size_t)ch * OSN * OSN;

    for (int i = tid; i < OSN * 4; i += 256) {
        const int yy = i >> 2, j = i & 3;
        async_load_b64(&bufA[j * OSN + yy], base + (size_t)yy * OSN + x0 + j);
    }
    wait_async0();
    __syncthreads();
    float2* res = fft768<DIR, true, true>(bufA, bufB, 4, tid, 256);
    for (int i = tid; i < OSN * 4; i += 256) {
        const int k = i >> 2, j = i & 3;
        async_store_b64(base + (size_t)k * OSN + x0 + j, &res[j * OSN + k]);
    }
}

// ---------------------------------------------------------------------------
// Pass 3: pixelwise subspace coupling Y[b,c] = sum_a K[b,a]*X[a,c] (complex),
// one wave per pixel, 8 v_wmma_f32_16x16x4_f32 per pixel, in place on grid.
// A frag (16x4 f32): lane<16 -> k=elem, lane>=16 -> k=2+elem; M = lane&15.
// B frag mirrors k mapping with N = lane&15 (ISA 7.12.2).
// ---------------------------------------------------------------------------
#define WMMA_F32X4(A, B, C) \
    __builtin_amdgcn_wmma_f32_16x16x4_f32(false, (A), false, (B), (short)0, (C), false, false)

__global__ __launch_bounds__(256) void einsum_wmma(const float* __restrict__ knr,
                                                   const float* __restrict__ kni,
                                                   float2* __restrict__ grid)
{
    const int wave = threadIdx.x >> 5;
    const int lane = threadIdx.x & 31;
    const int pix  = blockIdx.x * 8 + wave;
    const int y    = pix / OSN;
    const int x    = pix - y * OSN;
    const int row  = lane & 15;     // b (A rows) / c (B cols)
    const int hi   = lane >> 4;

    float ar[2][2], ai[2][2], br[2][2], bi[2][2];   // [k-chunk][elem]
#pragma unroll
    for (int ck = 0; ck < 2; ++ck) {
#pragma unroll
        for (int e = 0; e < 2; ++e) {
            const int k = ck * 4 + 2 * hi + e;      // contraction index a
            float kr = 0.f, kiv = 0.f, xr = 0.f, xiv = 0.f;
            if (k < NCOEFF) {
                if (row < NCOEFF) {
                    const size_t kidx = (((size_t)(row * NCOEFF + k)) * OSN + y) * OSN + x;
                    kr  = knr[kidx];
                    kiv = kni[kidx];
                }
                const float2 v = grid[(((size_t)(k * NCOIL + row)) * OSN + y) * OSN + x];
                xr = v.x; xiv = v.y;
            }
            ar[ck][e] = kr;  ai[ck][e] = kiv;
            br[ck][e] = xr;  bi[ck][e] = xiv;
        }
    }
    const v2f Kr0  = {ar[0][0], ar[0][1]},  Kr1  = {ar[1][0], ar[1][1]};
    const v2f Ki0  = {ai[0][0], ai[0][1]},  Ki1  = {ai[1][0], ai[1][1]};
    const v2f nKi0 = {-ai[0][0], -ai[0][1]}, nKi1 = {-ai[1][0], -ai[1][1]};
    const v2f Xr0  = {br[0][0], br[0][1]},  Xr1  = {br[1][0], br[1][1]};
    const v2f Xi0  = {bi[0][0], bi[0][1]},  Xi1  = {bi[1][0], bi[1][1]};

    v8f accR = {}, accI = {};
    accR = WMMA_F32X4(Kr0,  Xr0, accR);   // Yr = Kr*Xr - Ki*Xi
    accR = WMMA_F32X4(Kr1,  Xr1, accR);
    accR = WMMA_F32X4(nKi0, Xi0, accR);
    accR = WMMA_F32X4(nKi1, Xi1, accR);
    accI = WMMA_F32X4(Kr0,  Xi0, accI);   // Yi = Kr*Xi + Ki*Xr
    accI = WMMA_F32X4(Kr1,  Xi1, accI);
    accI = WMMA_F32X4(Ki0,  Xr0, accI);
    accI = WMMA_F32X4(Ki1,  Xr1, accI);

    // C/D layout: VGPR r, lanes 0-15 hold M=r, N=lane.  Only b=0..4 valid.
    if (hi == 0) {
#pragma unroll
        for (int b = 0; b < NCOEFF; ++b) {
            grid[(((size_t)(b * NCOIL + row)) * OSN + y) * OSN + x] =
                make_float2(accR[b], accI[b]);
        }
    }
}

// ---------------------------------------------------------------------------
// Pass 5: inverse row FFT only on the 384 cropped rows; crop x; write planes.
// Row staged global->LDS asynchronously; both centered signs folded in FFT.
// out layout [2][5][16][384][384]: real then imag.
// ---------------------------------------------------------------------------
__global__ __launch_bounds__(256) void inv_row_fft(const float2* __restrict__ grid,
                                                   float* __restrict__ out)
{
    __shared__ float2 bufA[OSN];
    __shared__ float2 bufB[OSN];
    const int yy  = blockIdx.x;          // 0..383
    const int ch  = blockIdx.y;
    const int tid = threadIdx.x;
    const float2* row = grid + (size_t)(ch * OSN + (yy + PAD)) * OSN;

    for (int k = tid; k < OSN; k += 256)
        async_load_b64(&bufA[k], &row[k]);
    wait_async0();
    __syncthreads();
    float2* res = fft768<1, true, true>(bufA, bufB, 1, tid, 256);

    const size_t obase    = ((size_t)ch * NN + yy) * NN;
    const size_t imag_off = (size_t)NCH * NN * NN;
    for (int x = PAD + tid; x < PAD + NN; x += 256) {
        float2 v = res[x];               // sign & scale already folded
        out[obase + (x - PAD)]            = v.x;
        out[obase + imag_off + (x - PAD)] = v.y;
    }
}

// ---------------------------------------------------------------------------
extern "C" void kernel_launch(void* const* d_in, const int* in_sizes, int n_in,
                              void* d_out, int out_size, void* d_ws, size_t ws_size,
                              hipStream_t stream)
{
    (void)in_sizes; (void)n_in; (void)out_size; (void)ws_size;
    const float* imr = (const float*)d_in[0];
    const float* imi = (const float*)d_in[1];
    const float* knr = (const float*)d_in[2];
    const float* kni = (const float*)d_in[3];
    float*  out  = (float*)d_out;
    float2* grid = (float2*)d_ws;    // [80][768][768] complex = 377 MB

    fwd_row_fft<<<dim3(OSN, NCH), 256, 0, stream>>>(imr, imi, grid);
    col_fft<-1><<<dim3(OSN / 4, NCH), 256, 0, stream>>>(grid);
    einsum_wmma<<<dim3((OSN * OSN) / 8), 256, 0, stream>>>(knr, kni, grid);
    col_fft<1><<<dim3(OSN / 4, NCH), 256, 0, stream>>>(grid);
    inv_row_fft<<<dim3(NN, NCH), 256, 0, stream>>>(grid, out);
}